// SpectrumEncoding_84164179132426
// MI455X (gfx1250) — compile-verified
//
#include <hip/hip_runtime.h>
#include <hip/hip_bf16.h>

// SpectrumEncoding: out[b,:] = sum_n intensity[b,n] * pe[ceil(loc[b,n]*10), :]
// B=1024, N=500 peaks, D=512. pe table = 61.4 MB -> L2-resident (192 MB L2).
// Roofline: 1.05 GB of gathered row reads (L2-bound), 0.52 GFLOP (trivial),
// ~70 MB HBM traffic (~3 us at 23.3 TB/s). The kernel is an L2->LDS streaming
// problem, so we use the CDNA5 async global->LDS copy path (ASYNCcnt) with a
// TRIPLE-buffered row pipeline for latency hiding. WMMA is deliberately not
// used: each batch is an M=1 GEMV over gathered rows (no shared B tile), so a
// 16x16 WMMA would do 16x redundant work on a bandwidth-bound kernel.

#define BATCH   1024
#define NPEAK   500
#define DMODEL  512
#define RESO    10.0f
#define CHUNK   10                  // rows per async chunk; 500 = 50 chunks
#define NCHUNK  (NPEAK / CHUNK)
#define NBUF    3                   // triple buffer: prefetch distance = 2 chunks
#define TPB     128                 // 4 wave32 waves

typedef float f32x4 __attribute__((ext_vector_type(4)));
typedef int   v4i   __attribute__((ext_vector_type(4)));
typedef __attribute__((address_space(1))) v4i* g_v4i_ptr;   // global
typedef __attribute__((address_space(3))) v4i* l_v4i_ptr;   // LDS

__device__ __forceinline__ void async_copy_b128(const float* gsrc, float* ldst) {
#if __has_builtin(__builtin_amdgcn_global_load_async_to_lds_b128)
    __builtin_amdgcn_global_load_async_to_lds_b128(
        (g_v4i_ptr)gsrc, (l_v4i_ptr)ldst, /*imm offset=*/0, /*cpol=*/0);
#else
    unsigned lofs = (unsigned)(unsigned long long)(l_v4i_ptr)ldst;
    asm volatile("global_load_async_to_lds_b128 %0, %1, off"
                 :: "v"(lofs), "v"(gsrc) : "memory");
#endif
}

template <int N>
__device__ __forceinline__ void wait_asynccnt() {
    // Inline asm with memory clobber: compiler cannot hoist the subsequent
    // LDS reads above the counter wait.
    asm volatile("s_wait_asynccnt %0" :: "n"(N) : "memory");
}

__global__ __launch_bounds__(TPB)
void SpectrumEncoding_84164179132426_kernel(const float* __restrict__ loc,
                                            const float* __restrict__ inten,
                                            const float* __restrict__ pe,
                                            float* __restrict__ out) {
    __shared__ int   sIdx[NPEAK];                                 // 2 KB
    __shared__ float sInt[NPEAK];                                 // 2 KB
    __shared__ __align__(16) float sRows[NBUF * CHUNK * DMODEL];  // 60 KB

    const int b    = blockIdx.x;
    const int tid  = threadIdx.x;
    const int wave = tid >> 5;
    const int lane = tid & 31;
    // Thread-private contiguous float4 slice of the D=512 output row.
    // Wave w covers dims [128w, 128w+128) -> each wave async-copies exactly
    // the LDS bytes it later reads: no inter-wave sync in the main loop.
    const int d0 = wave * 128 + lane * 4;

    // Stage indices + intensities for this batch into LDS (one barrier total).
    for (int n = tid; n < NPEAK; n += TPB) {
        float l = loc[b * NPEAK + n];
        sIdx[n] = (int)ceilf(l * RESO);
        sInt[n] = inten[b * NPEAK + n];
    }
    __syncthreads();

    // Issue one chunk of async row copies into LDS buffer `bufsel`.
    auto issue_chunk = [&](int chunk, int bufsel) {
        const int base = chunk * CHUNK;
        for (int r = 0; r < CHUNK; ++r) {
            const long row = (long)sIdx[base + r];
            async_copy_b128(pe + row * DMODEL + d0,
                            &sRows[(bufsel * CHUNK + r) * DMODEL + d0]);
        }
    };

    f32x4 acc = {0.f, 0.f, 0.f, 0.f};

    // Prologue: fill two of the three buffers.
    issue_chunk(0, 0);
    issue_chunk(1, 1);

    int bufsel = 0;   // buffer holding the chunk being consumed
    for (int c = 0; c < NCHUNK; ++c) {
        if (c + 2 < NCHUNK) {
            // Issue chunk c+2 two chunks ahead; afterwards at most chunks
            // c, c+1, c+2 are in flight (30 ops). Async ops complete in
            // order, so <=2*CHUNK outstanding means chunk c has landed.
            int nb = bufsel + 2; if (nb >= NBUF) nb -= NBUF;
            issue_chunk(c + 2, nb);
            wait_asynccnt<2 * CHUNK>();
        } else if (c + 1 < NCHUNK) {
            wait_asynccnt<CHUNK>();      // only chunk c+1 may remain in flight
        } else {
            wait_asynccnt<0>();          // drain
        }

        // Consume chunk c from LDS (wave-private slice, no barrier needed).
        const int base = c * CHUNK;
        for (int r = 0; r < CHUNK; ++r) {
            const float w = sInt[base + r];                     // LDS broadcast
            const f32x4 v =
                *(const f32x4*)&sRows[(bufsel * CHUNK + r) * DMODEL + d0];
            acc.x = __builtin_fmaf(v.x, w, acc.x);
            acc.y = __builtin_fmaf(v.y, w, acc.y);
            acc.z = __builtin_fmaf(v.z, w, acc.z);
            acc.w = __builtin_fmaf(v.w, w, acc.w);
        }
        if (++bufsel == NBUF) bufsel = 0;
    }

    *(f32x4*)&out[b * DMODEL + d0] = acc;   // global_store_b128
}

extern "C" void kernel_launch(void* const* d_in, const int* in_sizes, int n_in,
                              void* d_out, int out_size, void* d_ws, size_t ws_size,
                              hipStream_t stream) {
    const float* loc   = (const float*)d_in[0];   // [1024, 500]
    const float* inten = (const float*)d_in[1];   // [1024, 500]
    const float* pe    = (const float*)d_in[2];   // [30001, 512]
    float* out         = (float*)d_out;           // [1024, 512]
    (void)in_sizes; (void)n_in; (void)out_size; (void)d_ws; (void)ws_size;

    SpectrumEncoding_84164179132426_kernel<<<BATCH, TPB, 0, stream>>>(
        loc, inten, pe, out);
}